// MultiScalePixelAdaptiveDilatedConv_47699906789723
// MI455X (gfx1250) — compile-verified
//
#include <hip/hip_runtime.h>

typedef __attribute__((ext_vector_type(2))) float v2f;
typedef __attribute__((ext_vector_type(8))) float v8f;

#define Bb 2
#define Cc 64
#define Hh 256
#define Ww 256
#define TAPS 9
#define NDIL 3
#define PIXT 64              // pixels per workgroup
#define NTHREADS 128         // 4 waves
#define WP_TAP 4096          // floats per tap in packed weights: 16 ksteps * 4 rowblocks * 32 lanes * 2
#define WP_MAT (TAPS * WP_TAP)   // 36864 floats = 64 rows x 576 K

// Chunk LDS layout (B-fragment native, bank-conflict-free):
//   element (k, col) -> chunk[(k>>1)*PAIR_STRIDE + 2*col + (k&1)]
// PAIR_STRIDE = 160 dwords: +32 pad puts the upper half-wave's b64 reads on
// banks 32..63 while the lower half-wave uses banks 0..31.
#define PAIR_STRIDE 160
#define CHUNK_FLOATS (32 * PAIR_STRIDE)   // 32 K-pairs -> 20 KB

// ---------------------------------------------------------------------------
// Prep kernel 1: pack Woff (54,64,3,3) -> WMMA A-fragment layout, rows padded to 64 with 0
// packed idx = (((tap*16 + ks)*4 + rb)*32 + lane)*2 + j
//   row = rb*16 + lane%16 ; ch = ks*4 + 2*(lane/16) + j ; (ky,kx) = tap/3, tap%3
// ---------------------------------------------------------------------------
__global__ void pack_woff_kernel(const float* __restrict__ Woff, float* __restrict__ wp) {
    int pidx = blockIdx.x * blockDim.x + threadIdx.x;
    if (pidx >= WP_MAT) return;
    int j    = pidx & 1;
    int lane = (pidx >> 1) & 31;
    int rb   = (pidx >> 6) & 3;
    int ks   = (pidx >> 8) & 15;
    int tap  = pidx >> 12;
    int row  = rb * 16 + (lane & 15);
    int ch   = ks * 4 + 2 * (lane >> 4) + j;
    int ky   = tap / 3, kx = tap % 3;
    float v = 0.0f;
    if (row < 54) v = Woff[((row * Cc + ch) * 3 + ky) * 3 + kx];
    wp[pidx] = v;
}

// ---------------------------------------------------------------------------
// Prep kernel 2: Weff[i] = Wfuse_i (64x64) @ Wd_i (64x576), packed into A-fragment layout
// ---------------------------------------------------------------------------
__global__ void build_weff_kernel(const float* __restrict__ Wd, const float* __restrict__ Wfuse,
                                  float* __restrict__ weff) {
    int pidx = blockIdx.x * blockDim.x + threadIdx.x;
    if (pidx >= NDIL * WP_MAT) return;
    int i = pidx / WP_MAT;
    int r = pidx % WP_MAT;
    int j    = r & 1;
    int lane = (r >> 1) & 31;
    int rb   = (r >> 6) & 3;
    int ks   = (r >> 8) & 15;
    int tap  = r >> 12;
    int o    = rb * 16 + (lane & 15);
    int ch   = ks * 4 + 2 * (lane >> 4) + j;
    int ky   = tap / 3, kx = tap % 3;
    float acc = 0.0f;
    #pragma unroll 8
    for (int m = 0; m < Cc; ++m) {
        float wf = Wfuse[o * (NDIL * Cc) + i * Cc + m];
        float wd = Wd[(((i * Cc + m) * Cc + ch) * 3 + ky) * 3 + kx];
        acc += wf * wd;
    }
    weff[pidx] = acc;
}

// ---------------------------------------------------------------------------
// 64x64 K-chunk GEMM step using V_WMMA_F32_16X16X4_F32.
// wp: packed A fragments for this tap (WP_TAP floats). s: LDS chunk in paired-K layout.
// Wave w owns output rows 16w..16w+15; acc[cb] = 16x16 tile for pixel block cb.
// B fragment: lane needs s[krow][col], s[krow+1][col] with krow = ks*4 + 2*lh (even),
// which in paired-K layout is one aligned 8-byte load.
// ---------------------------------------------------------------------------
__device__ __forceinline__ void gemm_chunk(const float* __restrict__ wp,
                                           const float* __restrict__ s,
                                           v8f acc[4], int lane, int w) {
    const int lh = lane >> 4;   // 0/1
    const int ll = lane & 15;
    #pragma unroll
    for (int ks = 0; ks < 16; ++ks) {
        v2f a = *(const v2f*)(wp + (((ks * 4 + w) * 32 + lane) << 1));
        const int kpair = ks * 2 + lh;           // (krow >> 1)
        #pragma unroll
        for (int cb = 0; cb < 4; ++cb) {
            const int col = cb * 16 + ll;
            v2f b = *(const v2f*)(s + kpair * PAIR_STRIDE + (col << 1));
            acc[cb] = __builtin_amdgcn_wmma_f32_16x16x4_f32(
                false, a, false, b, (short)0, acc[cb], false, false);
        }
    }
}

// ---------------------------------------------------------------------------
// Main fused kernel: offset conv -> bilinear deformable sampling -> Weff GEMM
// One workgroup = 64 consecutive pixels of one row of one image.
// ---------------------------------------------------------------------------
__global__ void __launch_bounds__(NTHREADS)
msdc_main_kernel(const float* __restrict__ xg, const float* __restrict__ boff,
                 const float* __restrict__ wp_off, const float* __restrict__ weff,
                 float* __restrict__ out) {
    __shared__ float chunk[CHUNK_FLOATS]; // 20 KB: K-chunk, paired-K layout
    __shared__ float offbuf[64 * PIXT];   // 16 KB: offset channels (54 used)
    __shared__ float metaW[4 * PIXT];     // bilinear corner weights (masked)
    __shared__ int   metaI[4 * PIXT];     // bilinear corner plane indices

    const int tid  = threadIdx.x;
    const int lane = tid & 31;
    const int w    = tid >> 5;
    const int lh   = lane >> 4;
    const int ll   = lane & 15;

    const int tile  = blockIdx.x;                // 0..2047
    const int rowId = tile >> 2;                 // b*H + y
    const int x0    = (tile & 3) * PIXT;
    const int b     = rowId / Hh;
    const int y     = rowId % Hh;

    v8f acc[4];
    #pragma unroll
    for (int cb = 0; cb < 4; ++cb) acc[cb] = (v8f){};

    // ---------------- Phase 1: offset conv (3x3, pad 1) via WMMA ----------------
    #pragma unroll 1
    for (int tap = 0; tap < TAPS; ++tap) {
        __syncthreads();
        const int ky = tap / 3 - 1, kx = tap % 3 - 1;
        const int yy = y + ky;
        const bool yok = (yy >= 0) && (yy < Hh);
        for (int idx = tid; idx < Cc * PIXT; idx += NTHREADS) {
            const int ch = idx >> 6, p = idx & 63;
            const int xx = x0 + p + kx;
            float v = 0.0f;
            if (yok && xx >= 0 && xx < Ww)
                v = xg[((b * Cc + ch) * Hh + yy) * Ww + xx];
            chunk[(ch >> 1) * PAIR_STRIDE + (p << 1) + (ch & 1)] = v;
        }
        if (tap + 1 < TAPS)
            __builtin_prefetch(wp_off + (tap + 1) * WP_TAP + tid * 32, 0, 1);
        __syncthreads();
        gemm_chunk(wp_off + tap * WP_TAP, chunk, acc, lane, w);
    }

    // write offsets (+bias) to LDS; rows 54..63 are don't-care
    __syncthreads();
    #pragma unroll
    for (int cb = 0; cb < 4; ++cb) {
        #pragma unroll
        for (int r = 0; r < 8; ++r) {
            const int row = 16 * w + r + 8 * lh;
            const int col = cb * 16 + ll;
            float v = acc[cb][r];
            if (row < 54) v += boff[row];
            offbuf[row * PIXT + col] = v;
        }
    }

    // ---------------- Phase 2: sample + accumulate Weff_i * S_i ----------------
    #pragma unroll
    for (int cb = 0; cb < 4; ++cb) acc[cb] = (v8f){};

    const int DIL[NDIL] = {1, 2, 4};
    #pragma unroll 1
    for (int i = 0; i < NDIL; ++i) {
        const int d = DIL[i];
        #pragma unroll 1
        for (int tap = 0; tap < TAPS; ++tap) {
            __syncthreads();
            // bilinear metadata per pixel (independent of channel)
            if (tid < PIXT) {
                const int p = tid;
                const float dy = offbuf[(i * 18 + 2 * tap) * PIXT + p];
                const float dx = offbuf[(i * 18 + 2 * tap + 1) * PIXT + p];
                const float ys = (float)y + (float)((tap / 3 - 1) * d) + dy;
                const float xs = (float)(x0 + p) + (float)((tap % 3 - 1) * d) + dx;
                const float y0f = floorf(ys), x0f = floorf(xs);
                const float wy1 = ys - y0f, wx1 = xs - x0f;
                const float wy0 = 1.0f - wy1, wx0 = 1.0f - wx1;
                const float vy0 = (y0f >= 0.0f && y0f <= (float)(Hh - 1)) ? 1.0f : 0.0f;
                const float vy1 = (y0f + 1.0f >= 0.0f && y0f + 1.0f <= (float)(Hh - 1)) ? 1.0f : 0.0f;
                const float vx0 = (x0f >= 0.0f && x0f <= (float)(Ww - 1)) ? 1.0f : 0.0f;
                const float vx1 = (x0f + 1.0f >= 0.0f && x0f + 1.0f <= (float)(Ww - 1)) ? 1.0f : 0.0f;
                int y0i = (int)y0f, x0i = (int)x0f;
                int y0c = min(max(y0i, 0), Hh - 1);
                int y1c = min(max(y0i + 1, 0), Hh - 1);
                int x0c = min(max(x0i, 0), Ww - 1);
                int x1c = min(max(x0i + 1, 0), Ww - 1);
                metaI[0 * PIXT + p] = y0c * Ww + x0c;
                metaI[1 * PIXT + p] = y0c * Ww + x1c;
                metaI[2 * PIXT + p] = y1c * Ww + x0c;
                metaI[3 * PIXT + p] = y1c * Ww + x1c;
                metaW[0 * PIXT + p] = wy0 * wx0 * vy0 * vx0;
                metaW[1 * PIXT + p] = wy0 * wx1 * vy0 * vx1;
                metaW[2 * PIXT + p] = wy1 * wx0 * vy1 * vx0;
                metaW[3 * PIXT + p] = wy1 * wx1 * vy1 * vx1;
            }
            __syncthreads();
            // gather-fill chunk: value depends on (ch, p), metadata only on p
            for (int idx = tid; idx < Cc * PIXT; idx += NTHREADS) {
                const int ch = idx >> 6, p = idx & 63;
                const float* plane = xg + (size_t)(b * Cc + ch) * (Hh * Ww);
                float v = metaW[0 * PIXT + p] * plane[metaI[0 * PIXT + p]]
                        + metaW[1 * PIXT + p] * plane[metaI[1 * PIXT + p]]
                        + metaW[2 * PIXT + p] * plane[metaI[2 * PIXT + p]]
                        + metaW[3 * PIXT + p] * plane[metaI[3 * PIXT + p]];
                chunk[(ch >> 1) * PAIR_STRIDE + (p << 1) + (ch & 1)] = v;
            }
            // prefetch next tap's packed weights (L2 -> near cache) while GEMM runs
            {
                const int nt = i * TAPS + tap + 1;
                if (nt < NDIL * TAPS)
                    __builtin_prefetch(weff + nt * WP_TAP + tid * 32, 0, 1);
            }
            __syncthreads();
            gemm_chunk(weff + (i * TAPS + tap) * WP_TAP, chunk, acc, lane, w);
        }
    }

    // ---------------- Store final 64ch x 64px tile ----------------
    #pragma unroll
    for (int cb = 0; cb < 4; ++cb) {
        #pragma unroll
        for (int r = 0; r < 8; ++r) {
            const int row = 16 * w + r + 8 * lh;
            const int col = cb * 16 + ll;
            out[((b * Cc + row) * Hh + y) * Ww + x0 + col] = acc[cb][r];
        }
    }
}

// ---------------------------------------------------------------------------
extern "C" void kernel_launch(void* const* d_in, const int* in_sizes, int n_in,
                              void* d_out, int out_size, void* d_ws, size_t ws_size,
                              hipStream_t stream) {
    const float* x     = (const float*)d_in[0];
    const float* Woff  = (const float*)d_in[1];
    const float* boff  = (const float*)d_in[2];
    const float* Wd    = (const float*)d_in[3];
    const float* Wfuse = (const float*)d_in[4];
    float* out = (float*)d_out;

    float* wp_off = (float*)d_ws;              // 36864 floats
    float* weff   = wp_off + WP_MAT;           // 3 * 36864 floats

    pack_woff_kernel<<<(WP_MAT + 255) / 256, 256, 0, stream>>>(Woff, wp_off);
    build_weff_kernel<<<(NDIL * WP_MAT + 255) / 256, 256, 0, stream>>>(Wd, Wfuse, weff);

    const int nTiles = Bb * Hh * (Ww / PIXT);  // 2048
    msdc_main_kernel<<<nTiles, NTHREADS, 0, stream>>>(x, boff, wp_off, weff, out);
}